// ECHO_6390911336490
// MI455X (gfx1250) — compile-verified
//
#include <hip/hip_runtime.h>
#include <hip/hip_bf16.h>

// Problem constants (reference: N=50000, E=1600000, C=8, n_bins=2)
#define CCH 8          // channels
#define NB  2          // n_bins
#define DS  21         // bins inside the disk for n_bins=2 (5x5 minus 4 corners)

// disk map for w=5 grid, computed arithmetically (matches reference dMap):
//   row 0 / row 4 have 3 valid cells, rows 1..3 have 5; corners -> 0
__device__ __forceinline__ int dmap5(int a, int b) {
    int v = 5 * a + b - 2 + (a == 0 ? 1 : 0) - (a == 4 ? 1 : 0);
    bool corner = ((a == 0) | (a == 4)) & ((b == 0) | (b == 4));
    return corner ? 0 : v;
}

__device__ __forceinline__ void atomAddF32(float* p, float v) {
    // relaxed, device-scope fp32 add -> native global_atomic_add_f32 (no return)
    __hip_atomic_fetch_add(p, v, __ATOMIC_RELAXED, __HIP_MEMORY_SCOPE_AGENT);
}

__global__ void zero_ws_kernel(float4* __restrict__ ws, int n4) {
    int t = blockIdx.x * blockDim.x + threadIdx.x;
    if (t < n4) ws[t] = make_float4(0.f, 0.f, 0.f, 0.f);
}

// one thread per (edge, channel); lanes t%8 == channel -> x gathers are 64B
// coalesced per 8-lane group; edge payload (edges/ln/wxp) is broadcast-cached.
// All index math fits in 32 bits (E*C = 12.8M, hist floats = 16.8M).
__global__ void scatter_kernel(const float2* __restrict__ x,     // [N*C] complex
                               const int2*   __restrict__ edges, // [E]
                               const float2* __restrict__ ln,    // [E] complex
                               const float2* __restrict__ wxp,   // [E] complex
                               float*        __restrict__ hist,  // [N*C*DS*2] interleaved
                               int total) {
    int t = blockIdx.x * blockDim.x + threadIdx.x;
    if (t >= total) return;
    int e = t >> 3;
    int c = t & 7;

    int2   ev = edges[e];
    float2 xv = x[ev.x * CCH + c];
    float2 lv = ln[e];
    float2 wv = wxp[e];

    float xr = xv.x, xi = xv.y;

    // unit_conj = conj(x)/|x|  (replaces atan2 + cos/sin with one rsqrt)
    float m2  = xr * xr + xi * xi;
    float inv = (m2 > 0.f) ? rsqrtf(m2) : 0.f; // m2==0 contributes zero anyway
    float ur = xr * inv;
    float ui = -xi * inv;

    // aligned = lnC * unit_conj ; p = aligned * n_bins
    float pr = (float)NB * (lv.x * ur - lv.y * ui);
    float pi = (float)NB * (lv.x * ui + lv.y * ur);

    const float nbf = (float)NB;
    float pCx = fminf(fmaxf(ceilf(pr),  -nbf), nbf);
    float pCy = fminf(fmaxf(ceilf(pi),  -nbf), nbf);
    float pFx = fminf(fmaxf(floorf(pr), -nbf), nbf);
    float pFy = fminf(fmaxf(floorf(pi), -nbf), nbf);

    float r0 = (pCx - pr) * (pCy - pi);
    float r1 = (pr - pFx) * (pi - pFy);
    float r2 = (pr - pFx) * (pCy - pi);
    float r3 = (pCx - pr) * (pi - pFy);

    int Fx = (int)pFx + NB, Fy = (int)pFy + NB;
    int Cx = (int)pCx + NB, Cy = (int)pCy + NB;
    int i0 = dmap5(Fx, Fy);
    int i1 = dmap5(Cx, Cy);
    int i2 = dmap5(Cx, Fy);
    int i3 = dmap5(Fx, Cy);

    float nz = ((xr != 0.f) || (xi != 0.f)) ? 1.f : 0.f;
    float wR = (xr * wv.x - xi * wv.y) * nz;
    float wI = (xr * wv.y + xi * wv.x) * nz;

    float* bin = hist + 2 * ((ev.y * CCH + c) * DS);

    atomAddF32(bin + 2 * i0,     wR * r0);
    atomAddF32(bin + 2 * i0 + 1, wI * r0);
    atomAddF32(bin + 2 * i1,     wR * r1);
    atomAddF32(bin + 2 * i1 + 1, wI * r1);
    atomAddF32(bin + 2 * i2,     wR * r2);
    atomAddF32(bin + 2 * i2 + 1, wI * r2);
    atomAddF32(bin + 2 * i3,     wR * r3);
    atomAddF32(bin + 2 * i3 + 1, wI * r3);
}

__global__ void finalize_kernel(const float2* __restrict__ hist, // interleaved complex
                                float* __restrict__ out, int n) {
    int t = blockIdx.x * blockDim.x + threadIdx.x;
    if (t < n) {
        float2 h = hist[t];
        out[t] = sqrtf(h.x * h.x + h.y * h.y + 1e-12f);
    }
}

extern "C" void kernel_launch(void* const* d_in, const int* in_sizes, int n_in,
                              void* d_out, int out_size, void* d_ws, size_t ws_size,
                              hipStream_t stream) {
    const float2* x     = (const float2*)d_in[0]; // N*C complex
    const int2*   edges = (const int2*)  d_in[1]; // E pairs
    const float2* ln    = (const float2*)d_in[2]; // E complex
    const float2* wxp   = (const float2*)d_in[3]; // E complex
    // d_in[4] = n_bins (==2, baked into NB)

    const int N = in_sizes[0] / (CCH * 2);
    const int E = in_sizes[1] / 2;

    float* hist = (float*)d_ws;              // N*C*DS complex, interleaved
    const int nOut  = N * CCH * DS;          // 8.4M
    const int histF = 2 * nOut;              // 16.8M floats, multiple of 4

    // 1) zero accumulator
    {
        int n4 = histF / 4;
        int blocks = (n4 + 255) / 256;
        zero_ws_kernel<<<blocks, 256, 0, stream>>>((float4*)hist, n4);
    }

    // 2) scatter-add (one thread per edge-channel)
    {
        int total = E * CCH;
        int blocks = (total + 255) / 256;
        scatter_kernel<<<blocks, 256, 0, stream>>>(x, edges, ln, wxp, hist, total);
    }

    // 3) soft_abs finalize
    {
        int blocks = (nOut + 255) / 256;
        finalize_kernel<<<blocks, 256, 0, stream>>>((const float2*)hist, (float*)d_out, nOut);
    }
}